// CapsuleLoss_7739531067460
// MI455X (gfx1250) — compile-verified
//
#include <hip/hip_runtime.h>
#include <math.h>

// CapsuleLoss for MI455X (gfx1250, wave32).
// labels scan (8 MiB) is the only real work: ~0.36us at 23.3 TB/s.
// Presence per image fits a 32-bit mask (C=21). Caps lengths via
// V_WMMA_F32_16X16X4_F32 (diag of X*X^T), full fp32 precision.

#define BATCH   8
#define NCLS    21
#define PIX     (512 * 512)
#define M_PLUS  0.9f
#define M_MINUS 0.1f
#define LAM     0.5f

typedef __attribute__((ext_vector_type(2))) float v2f;
typedef __attribute__((ext_vector_type(8))) float v8f;

// ---------------------------------------------------------------- init ------
__global__ void init_masks_kernel(unsigned* __restrict__ masks) {
    if (threadIdx.x < BATCH) masks[threadIdx.x] = 0u;
}

// ---------------------------------------------------------------- scan ------
// grid = (128, BATCH), block = 256 (8 waves). Each thread: 2 x int4 loads
// (global_load_b128), builds a class-presence bitmask. Valid labels are
// 0..20; 255 (ignore) maps to no bit. OR is commutative -> deterministic.
__global__ __launch_bounds__(256) void scan_labels_kernel(
    const int* __restrict__ labels, unsigned* __restrict__ masks) {
    __shared__ unsigned smask;
    if (threadIdx.x == 0) smask = 0u;
    __syncthreads();

    const int b = blockIdx.y;
    const int4* __restrict__ p =
        reinterpret_cast<const int4*>(labels) + (size_t)b * (PIX / 4);
    const int nvec = PIX / 4;

    unsigned local = 0u;
    for (int i = blockIdx.x * blockDim.x + threadIdx.x; i < nvec;
         i += gridDim.x * blockDim.x) {
        int4 v = p[i];
        local |= ((unsigned)v.x < (unsigned)NCLS) ? (1u << v.x) : 0u;
        local |= ((unsigned)v.y < (unsigned)NCLS) ? (1u << v.y) : 0u;
        local |= ((unsigned)v.z < (unsigned)NCLS) ? (1u << v.z) : 0u;
        local |= ((unsigned)v.w < (unsigned)NCLS) ? (1u << v.w) : 0u;
    }

    // wave32 OR-reduce
    #pragma unroll
    for (int off = 16; off > 0; off >>= 1)
        local |= (unsigned)__shfl_xor((int)local, off, 32);

    if ((threadIdx.x & 31) == 0 && local) atomicOr(&smask, local);
    __syncthreads();
    if (threadIdx.x == 0 && smask) atomicOr(&masks[b], smask);
}

// ------------------------------------------------------------- finalize -----
// One block, 8 waves. Wave w computes batch w's 21 capsule lengths as the
// diagonal of X * X^T using V_WMMA_F32_16X16X4_F32 (two 16x16 tiles, four
// K=4 steps each). Then margin + block reduction -> scalar loss.
__global__ __launch_bounds__(256) void finalize_kernel(
    const float* __restrict__ caps, const unsigned* __restrict__ masks,
    float* __restrict__ out) {
    __shared__ float lens_s[BATCH][NCLS];
    __shared__ float red[256];

    const int tid  = threadIdx.x;
    const int lane = tid & 31;
    const int b    = tid >> 5;  // wave id == batch id (8 waves)

#if __has_builtin(__builtin_amdgcn_wmma_f32_16x16x4_f32)
    {
        const float* __restrict__ X = caps + (size_t)b * NCLS * 16;
        #pragma unroll
        for (int t = 0; t < 2; ++t) {
            v8f acc = {};
            // f32 16x4 A layout: lanes 0-15 -> M=lane, K = k0,k0+1 ;
            //                    lanes 16-31 -> M=lane-16, K = k0+2,k0+3.
            const int  m     = lane & 15;
            const int  row   = t * 16 + m;
            const bool rowok = (row < NCLS);
            #pragma unroll
            for (int k0 = 0; k0 < 16; k0 += 4) {
                const int kk = (lane < 16) ? k0 : (k0 + 2);
                v2f a;
                a.x = rowok ? X[row * 16 + kk]     : 0.0f;
                a.y = rowok ? X[row * 16 + kk + 1] : 0.0f;
                // A == B per-lane for X*X^T under the documented striping.
                acc = __builtin_amdgcn_wmma_f32_16x16x4_f32(
                    false, a, false, a, (short)0, acc, false, false);
            }
            // 16x16 f32 C/D layout: lane 0-15 holds (M=vgpr, N=lane);
            // lane 16-31 holds (M=8+vgpr, N=lane-16). Diagonal lives in
            // lanes 0-7 (vgpr=lane) and lanes 24-31 (vgpr=lane-24).
            const int sel = (lane < 8) ? lane : ((lane >= 24) ? (lane - 24) : -1);
            float pick = 0.0f;
            #pragma unroll
            for (int j = 0; j < 8; ++j)
                if (sel == j) pick = acc[j];
            if (sel >= 0) {
                const int drow = t * 16 + ((lane < 8) ? lane : (8 + lane - 24));
                if (drow < NCLS) lens_s[b][drow] = sqrtf(pick);
            }
        }
    }
#else
    // scalar fallback (also what the host pass parses)
    if (tid < BATCH * NCLS) {
        const int bb = tid / NCLS, cc = tid % NCLS;
        const float* __restrict__ X = caps + ((size_t)bb * NCLS + cc) * 16;
        float s = 0.0f;
        #pragma unroll
        for (int k = 0; k < 16; ++k) s += X[k] * X[k];
        lens_s[bb][cc] = sqrtf(s);
    }
#endif
    __syncthreads();

    float e = 0.0f;
    if (tid < BATCH * NCLS) {
        const int bb = tid / NCLS, cc = tid % NCLS;
        const float L       = lens_s[bb][cc];
        const float present = (float)((masks[bb] >> cc) & 1u);
        const float left    = fmaxf(M_PLUS - L, 0.0f);
        const float right   = fmaxf(L - M_MINUS, 0.0f);
        e = present * left * left + LAM * (1.0f - present) * right * right;
    }
    red[tid] = e;
    __syncthreads();
    #pragma unroll
    for (int s = 128; s > 0; s >>= 1) {
        if (tid < s) red[tid] += red[tid + s];
        __syncthreads();
    }
    if (tid == 0) out[0] = red[0] * (1.0f / BATCH);
}

// --------------------------------------------------------------- launch -----
extern "C" void kernel_launch(void* const* d_in, const int* in_sizes, int n_in,
                              void* d_out, int out_size, void* d_ws,
                              size_t ws_size, hipStream_t stream) {
    (void)in_sizes; (void)n_in; (void)out_size; (void)ws_size;
    const float* caps   = (const float*)d_in[0];   // [8,21,16] f32
    const int*   labels = (const int*)d_in[1];     // [8,512,512] i32
    float*       out    = (float*)d_out;           // scalar
    unsigned*    masks  = (unsigned*)d_ws;         // 8 words of scratch

    init_masks_kernel<<<1, 32, 0, stream>>>(masks);
    dim3 grid(128, BATCH);
    scan_labels_kernel<<<grid, 256, 0, stream>>>(labels, masks);
    finalize_kernel<<<1, 256, 0, stream>>>(caps, masks, out);
}